// KANLayer_77936476553397
// MI455X (gfx1250) — compile-verified
//
#include <hip/hip_runtime.h>
#include <hip/hip_bf16.h>

#define KBATCH 2048
#define KCIN   128
#define KCOUT  128

typedef __attribute__((ext_vector_type(2))) float v2f;
typedef __attribute__((ext_vector_type(8))) float v8f;

// Build the degree-3 extended knot vector exactly like the reference:
// h = (g[5]-g[0])/5, extend 3 knots each side by sequential +-h.
__device__ __forceinline__ void kan_knots(const float* __restrict__ grid_x, float t[12]) {
  float g[6];
#pragma unroll
  for (int j = 0; j < 6; ++j) g[j] = grid_x[j];
  const float h = (g[5] - g[0]) / 5.0f;
#pragma unroll
  for (int j = 0; j < 6; ++j) t[3 + j] = g[j];
  t[2] = t[3] - h;  t[1] = t[2] - h;  t[0] = t[1] - h;
  t[9] = t[8] + h;  t[10] = t[9] + h; t[11] = t[10] + h;
}

// Cox-de Boor recursion, degree 3, 8 basis functions (matches reference b_spline).
__device__ __forceinline__ void kan_basis(float x, const float t[12], float B[8]) {
  float B0[11];
#pragma unroll
  for (int j = 0; j < 11; ++j)
    B0[j] = (x >= t[j] && x < t[j + 1]) ? 1.0f : 0.0f;
  float B1[10];
#pragma unroll
  for (int j = 0; j < 10; ++j)
    B1[j] = (x - t[j]) / (t[j + 1] - t[j]) * B0[j]
          + (t[j + 2] - x) / (t[j + 2] - t[j + 1]) * B0[j + 1];
  float B2[9];
#pragma unroll
  for (int j = 0; j < 9; ++j)
    B2[j] = (x - t[j]) / (t[j + 2] - t[j]) * B1[j]
          + (t[j + 3] - x) / (t[j + 3] - t[j + 1]) * B1[j + 1];
#pragma unroll
  for (int j = 0; j < 8; ++j)
    B[j]  = (x - t[j]) / (t[j + 3] - t[j]) * B2[j]
          + (t[j + 4] - x) / (t[j + 4] - t[j + 1]) * B2[j + 1];
}

// ---------------------------------------------------------------------------
// Kernel 1: bandwidth kernel. Writes preacts / postacts / postspline (403 MB)
// with fully coalesced float4 stores. Basis computed once per (b,i) in LDS,
// reused across all 128 outputs; coef rows hoisted out of the batch loop.
// ---------------------------------------------------------------------------
__global__ __launch_bounds__(256) void kan_elem(
    const float* __restrict__ x, const float* __restrict__ grid_x,
    const float* __restrict__ coef, const float* __restrict__ sbase,
    const float* __restrict__ ssp,
    float* __restrict__ preacts, float* __restrict__ postacts,
    float* __restrict__ postspline)
{
  __shared__ __align__(16) float sBas[8 * 128 * 8];  // [b][i][g]  32 KB
  __shared__ __align__(16) float sX  [8 * 128];      //             4 KB
  __shared__ __align__(16) float sSil[8 * 128];      //             4 KB

  const int tid = threadIdx.x;
  const int b0  = blockIdx.x * 8;

  float t[12];
  kan_knots(grid_x, t);

  for (int p = tid; p < 8 * 128; p += 256) {
    const int bb = p >> 7, i = p & 127;
    const float xv = x[(size_t)(b0 + bb) * KCIN + i];
    float bas[8];
    kan_basis(xv, t, bas);
#pragma unroll
    for (int g = 0; g < 8; ++g) sBas[(size_t)p * 8 + g] = bas[g];
    sX[p]   = xv;
    sSil[p] = xv / (1.0f + expf(-xv));
  }
  __syncthreads();

  const int lane = tid & 31;
  const int wave = tid >> 5;
  const int i0   = lane * 4;       // lane owns 4 consecutive i -> coalesced stores
  const int ob   = wave * 16;      // wave owns a 16-wide o strip

  for (int oo = 0; oo < 16; ++oo) {
    const int o  = ob + oo;
    const int nb = o * KCIN + i0;
    float4 cA[4], cB[4];
#pragma unroll
    for (int j = 0; j < 4; ++j) {
      const float4* cr = (const float4*)&coef[(size_t)(nb + j) * 8];
      cA[j] = cr[0];
      cB[j] = cr[1];
    }
    const float4 sb4 = *(const float4*)&sbase[nb];
    const float4 ss4 = *(const float4*)&ssp[nb];
    for (int bb = 0; bb < 8; ++bb) {
      const float4 x4 = *(const float4*)&sX  [bb * 128 + i0];
      const float4 s4 = *(const float4*)&sSil[bb * 128 + i0];
      float y[4];
#pragma unroll
      for (int j = 0; j < 4; ++j) {
        const float4* br = (const float4*)&sBas[(size_t)(bb * 128 + i0 + j) * 8];
        const float4 bl = br[0], bh = br[1];
        y[j] = cA[j].x * bl.x + cA[j].y * bl.y + cA[j].z * bl.z + cA[j].w * bl.w
             + cB[j].x * bh.x + cB[j].y * bh.y + cB[j].z * bh.z + cB[j].w * bh.w;
      }
      const size_t base = ((size_t)(b0 + bb) * KCOUT + o) * KCIN + i0;
      const float4 y4 = make_float4(y[0], y[1], y[2], y[3]);
      const float4 pa = make_float4(sb4.x * s4.x + ss4.x * y4.x,
                                    sb4.y * s4.y + ss4.y * y4.y,
                                    sb4.z * s4.z + ss4.z * y4.z,
                                    sb4.w * s4.w + ss4.w * y4.w);
      *(float4*)&preacts[base]    = x4;
      *(float4*)&postacts[base]   = pa;
      *(float4*)&postspline[base] = y4;
    }
  }
}

// ---------------------------------------------------------------------------
// Kernel 2: out[b,o] via f32 WMMA GEMM. out = silu(x) @ sb^T + Bas @ (ss*coef)^T
// M=16 batches/block, N=16 outs/wave (8 waves), K = 128 (silu) + 1024 (spline).
// A-panel lives in LDS (64 KB), B-operand generated on the fly from L2.
// ---------------------------------------------------------------------------
__global__ __launch_bounds__(256) void kan_out_wmma(
    const float* __restrict__ x, const float* __restrict__ grid_x,
    const float* __restrict__ coef, const float* __restrict__ sbase,
    const float* __restrict__ ssp, float* __restrict__ out)
{
  __shared__ __align__(16) float panel[16 * 1024];  // 64 KB A-panel

  const int tid  = threadIdx.x;
  const int lane = tid & 31;
  const int wave = tid >> 5;
  const int b0   = blockIdx.x * 16;

  float t[12];
  kan_knots(grid_x, t);

  // ---- pass 1 fill: silu panel, A[b][i] (first 8 KB of panel) ----
  for (int p = tid; p < 16 * 128; p += 256) {
    const int bb = p >> 7, i = p & 127;
    const float xv = x[(size_t)(b0 + bb) * KCIN + i];
    panel[bb * 128 + i] = xv / (1.0f + expf(-xv));
  }
  __syncthreads();

  v8f c = {};
  const int m  = lane & 15;        // A row (batch) and B col (out) index
  const int kh = (lane >> 4) * 2;  // K offset: lanes 0-15 -> K{0,1}, 16-31 -> K{2,3}
  const int o  = wave * 16 + m;

  // ---- GEMM 1: K = 128, W[i][o] = sbase[o*128+i] ----
  for (int k0 = 0; k0 < 128; k0 += 4) {
    const int k = k0 + kh;
    v2f a, b;
    a.x = panel[m * 128 + k];
    a.y = panel[m * 128 + k + 1];
    b.x = sbase[(size_t)o * KCIN + k];
    b.y = sbase[(size_t)o * KCIN + k + 1];
    c = __builtin_amdgcn_wmma_f32_16x16x4_f32(false, a, false, b, (short)0, c,
                                              false, false);
  }
  __syncthreads();

  // ---- pass 2 fill: spline panel, A[b][g*128+i] = B_g(x[b,i]) ----
  for (int p = tid; p < 16 * 128; p += 256) {
    const int bb = p >> 7, i = p & 127;
    const float xv = x[(size_t)(b0 + bb) * KCIN + i];
    float bas[8];
    kan_basis(xv, t, bas);
#pragma unroll
    for (int g = 0; g < 8; ++g) panel[bb * 1024 + g * 128 + i] = bas[g];
  }
  __syncthreads();

  // ---- GEMM 2: K = 1024, W[g*128+i][o] = ssp[n]*coef[n][g], n = o*128+i ----
  for (int k0 = 0; k0 < 1024; k0 += 4) {
    const int k = k0 + kh;                 // even, never crosses an i=128 boundary
    const int g = k >> 7;
    const int i = k & 127;
    const int n = o * KCIN + i;
    v2f a, b;
    a.x = panel[m * 1024 + k];
    a.y = panel[m * 1024 + k + 1];
    b.x = ssp[n]     * coef[(size_t)n * 8 + g];
    b.y = ssp[n + 1] * coef[(size_t)(n + 1) * 8 + g];
    c = __builtin_amdgcn_wmma_f32_16x16x4_f32(false, a, false, b, (short)0, c,
                                              false, false);
  }

  // ---- store C tile: VGPR r -> rows r (lanes 0-15) / r+8 (lanes 16-31) ----
  const int bhalf = (lane >> 4) * 8;
#pragma unroll
  for (int r = 0; r < 8; ++r) {
    const int bg = b0 + bhalf + r;
    out[(size_t)bg * KCOUT + o] = c[r];
  }
}

extern "C" void kernel_launch(void* const* d_in, const int* in_sizes, int n_in,
                              void* d_out, int out_size, void* d_ws, size_t ws_size,
                              hipStream_t stream) {
  const float* x      = (const float*)d_in[0];
  const float* grid_x = (const float*)d_in[1];
  const float* coef   = (const float*)d_in[2];
  const float* sbase  = (const float*)d_in[3];
  const float* ssp    = (const float*)d_in[4];

  float* out        = (float*)d_out;                            // (2048,128)
  float* preacts    = out + (size_t)KBATCH * KCOUT;             // (2048,128,128)
  float* postacts   = preacts + (size_t)KBATCH * KCOUT * KCIN;  // (2048,128,128)
  float* postspline = postacts + (size_t)KBATCH * KCOUT * KCIN; // (2048,128,128)

  kan_elem<<<KBATCH / 8, 256, 0, stream>>>(x, grid_x, coef, sbase, ssp,
                                           preacts, postacts, postspline);
  kan_out_wmma<<<KBATCH / 16, 256, 0, stream>>>(x, grid_x, coef, sbase, ssp, out);
}